// CrossAttention_31044023615565
// MI455X (gfx1250) — compile-verified
//
#include <hip/hip_runtime.h>

typedef __attribute__((ext_vector_type(2))) float v2f;
typedef __attribute__((ext_vector_type(8))) float v8f;

#define NUM_HEADS 4
#define HEAD_DIM 32
#define QUERY_DIM 256
#define KEY_DIM 128
#define INNER 128
#define NTOK 1600   // H*W = 40*40
#define NBT 8       // B*T

__device__ __forceinline__ v8f wmma4(v2f a, v2f b, v8f c) {
  // D = A(16x4 f32) * B(4x16 f32) + C ; full fp32 WMMA on CDNA5
  return __builtin_amdgcn_wmma_f32_16x16x4_f32(false, a, false, b, (short)0, c,
                                               false, false);
}

// dst[bt][n][i] = sum_d src[bt][d][n] * W[d][i]
// src: [NBT][D][NTOK] (row = d), W: [D][INNER], dst: [NBT][NTOK][INNER]
__global__ __launch_bounds__(32)
void proj_kernel(const float* __restrict__ src, const float* __restrict__ W,
                 float* __restrict__ dst, int D) {
  const int lane = threadIdx.x;
  const int m  = lane & 15;        // A row / B col / D col
  const int hi = lane >> 4;        // half-wave selector
  const int kb = hi << 1;          // k-pair base within a 4-wide k step
  const int n0 = blockIdx.x * 16;
  const int bt = blockIdx.y;

  const float* sb = src + (size_t)bt * D * NTOK + n0 + m;
  float* db = dst + (size_t)bt * NTOK * INNER;

  v8f acc[INNER / 16];
#pragma unroll
  for (int it = 0; it < INNER / 16; ++it) acc[it] = {};

  for (int k = 0; k < D; k += 4) {
    v2f a;
    a.x = sb[(size_t)(k + kb) * NTOK];
    a.y = sb[(size_t)(k + kb + 1) * NTOK];
    const float* wb = W + (size_t)(k + kb) * INNER + m;
#pragma unroll
    for (int it = 0; it < INNER / 16; ++it) {
      v2f b;
      b.x = wb[it * 16];
      b.y = wb[INNER + it * 16];
      acc[it] = wmma4(a, b, acc[it]);
    }
  }
#pragma unroll
  for (int it = 0; it < INNER / 16; ++it)
#pragma unroll
    for (int r = 0; r < 8; ++r)
      db[(size_t)(n0 + r + hi * 8) * INNER + it * 16 + m] = acc[it][r];
}

// Flash-style attention for one 16-row query tile of one head of one (b,t).
// Q,K,V,O: [NBT][NTOK][INNER]
__global__ __launch_bounds__(32)
void attn_kernel(const float* __restrict__ Q, const float* __restrict__ K,
                 const float* __restrict__ V, float* __restrict__ O) {
  __shared__ float pls[16 * 17];   // P tile, padded to dodge bank conflicts
  const int lane = threadIdx.x;
  const int m  = lane & 15;
  const int hi = lane >> 4;
  const int kb = hi << 1;
  const int q0 = blockIdx.x * 16;
  const int h  = blockIdx.y;
  const int bt = blockIdx.z;
  const float scale = 0.17677669529663687f;  // 1/sqrt(HEAD_DIM)

  const size_t btoff = (size_t)bt * NTOK * INNER;
  const float* Qb  = Q + btoff + (size_t)q0 * INNER + h * HEAD_DIM;
  const float* Kb0 = K + btoff + h * HEAD_DIM;
  const float* Vb0 = V + btoff + h * HEAD_DIM;

  // Q tile in A-fragments: 8 k-steps cover HEAD_DIM=32
  v2f aq[8];
#pragma unroll
  for (int j = 0; j < 8; ++j) {
    aq[j].x = Qb[m * INNER + 4 * j + kb];
    aq[j].y = Qb[m * INNER + 4 * j + kb + 1];
  }

  // ---- pass 1: row max over all 1600 keys ----
  float rm[8];
#pragma unroll
  for (int r = 0; r < 8; ++r) rm[r] = -3.0e30f;

  for (int t = 0; t < NTOK / 16; ++t) {
    const float* Kb = Kb0 + (size_t)t * 16 * INNER + m * INNER;
    v8f s = {};
#pragma unroll
    for (int j = 0; j < 8; ++j) {
      v2f b;
      b.x = Kb[4 * j + kb];
      b.y = Kb[4 * j + kb + 1];
      s = wmma4(aq[j], b, s);
    }
#pragma unroll
    for (int r = 0; r < 8; ++r) rm[r] = fmaxf(rm[r], s[r] * scale);
  }
#pragma unroll
  for (int r = 0; r < 8; ++r) {
    rm[r] = fmaxf(rm[r], __shfl_xor(rm[r], 1, 32));
    rm[r] = fmaxf(rm[r], __shfl_xor(rm[r], 2, 32));
    rm[r] = fmaxf(rm[r], __shfl_xor(rm[r], 4, 32));
    rm[r] = fmaxf(rm[r], __shfl_xor(rm[r], 8, 32));
  }

  // ---- pass 2: recompute scores, exp, accumulate P*V ----
  float rs[8];
#pragma unroll
  for (int r = 0; r < 8; ++r) rs[r] = 0.0f;
  v8f o0 = {}, o1 = {};

  for (int t = 0; t < NTOK / 16; ++t) {
    const float* Kb = Kb0 + (size_t)t * 16 * INNER + m * INNER;
    v8f s = {};
#pragma unroll
    for (int j = 0; j < 8; ++j) {
      v2f b;
      b.x = Kb[4 * j + kb];
      b.y = Kb[4 * j + kb + 1];
      s = wmma4(aq[j], b, s);
    }
#pragma unroll
    for (int r = 0; r < 8; ++r) {
      float p = __expf(s[r] * scale - rm[r]);
      rs[r] += p;
      pls[(r + hi * 8) * 17 + m] = p;   // D-layout -> LDS row-major
    }
    __syncthreads();

    const float* Vb = Vb0 + (size_t)t * 16 * INNER;
#pragma unroll
    for (int jj = 0; jj < 4; ++jj) {
      v2f ap;
      ap.x = pls[m * 17 + 4 * jj + kb];  // A-layout read of P
      ap.y = pls[m * 17 + 4 * jj + kb + 1];
      const int k0 = 4 * jj + kb;
      v2f b0, b1;
      b0.x = Vb[k0 * INNER + m];
      b0.y = Vb[(k0 + 1) * INNER + m];
      o0 = wmma4(ap, b0, o0);
      b1.x = Vb[k0 * INNER + 16 + m];
      b1.y = Vb[(k0 + 1) * INNER + 16 + m];
      o1 = wmma4(ap, b1, o1);
    }
    __syncthreads();
  }

#pragma unroll
  for (int r = 0; r < 8; ++r) {
    rs[r] += __shfl_xor(rs[r], 1, 32);
    rs[r] += __shfl_xor(rs[r], 2, 32);
    rs[r] += __shfl_xor(rs[r], 4, 32);
    rs[r] += __shfl_xor(rs[r], 8, 32);
  }

  float* Ob = O + btoff + h * HEAD_DIM;
#pragma unroll
  for (int r = 0; r < 8; ++r) {
    const float inv = 1.0f / rs[r];
    const size_t row = (size_t)(q0 + r + hi * 8) * INNER;
    Ob[row + m]      = o0[r] * inv;
    Ob[row + 16 + m] = o1[r] * inv;
  }
}

// out[bt][o][n] = sum_i A[bt][n][i] * Wo[i][o] + bo[o]   (stores transposed)
__global__ __launch_bounds__(32)
void outproj_kernel(const float* __restrict__ A, const float* __restrict__ Wo,
                    const float* __restrict__ bo, float* __restrict__ out) {
  const int lane = threadIdx.x;
  const int m  = lane & 15;
  const int hi = lane >> 4;
  const int kb = hi << 1;
  const int n0 = blockIdx.x * 16;
  const int o0 = blockIdx.y * 16;
  const int bt = blockIdx.z;

  const float* Ab = A + (size_t)bt * NTOK * INNER + (size_t)(n0 + m) * INNER;
  v8f acc = {};
  for (int k = 0; k < INNER; k += 4) {
    v2f a;
    a.x = Ab[k + kb];
    a.y = Ab[k + kb + 1];
    v2f b;
    b.x = Wo[(size_t)(k + kb) * QUERY_DIM + o0 + m];
    b.y = Wo[(size_t)(k + kb + 1) * QUERY_DIM + o0 + m];
    acc = wmma4(a, b, acc);
  }
  const float bias = bo[o0 + m];
  float* ob = out + (size_t)bt * QUERY_DIM * NTOK + (size_t)(o0 + m) * NTOK + n0;
#pragma unroll
  for (int r = 0; r < 8; ++r) ob[r + hi * 8] = acc[r] + bias;
}

extern "C" void kernel_launch(void* const* d_in, const int* in_sizes, int n_in,
                              void* d_out, int out_size, void* d_ws, size_t ws_size,
                              hipStream_t stream) {
  const float* query = (const float*)d_in[0];
  const float* key   = (const float*)d_in[1];
  const float* value = (const float*)d_in[2];
  const float* Wq = (const float*)d_in[3];
  const float* Wk = (const float*)d_in[4];
  const float* Wv = (const float*)d_in[5];
  const float* Wo = (const float*)d_in[6];
  const float* bo = (const float*)d_in[7];
  float* out = (float*)d_out;

  const size_t mat = (size_t)NBT * NTOK * INNER;  // 1638400 floats
  float* Qw = (float*)d_ws;
  float* Kw = Qw + mat;
  float* Vw = Kw + mat;
  float* Aw = Vw + mat;

  dim3 blk(32);
  proj_kernel<<<dim3(NTOK / 16, NBT), blk, 0, stream>>>(query, Wq, Qw, QUERY_DIM);
  proj_kernel<<<dim3(NTOK / 16, NBT), blk, 0, stream>>>(key,   Wk, Kw, KEY_DIM);
  proj_kernel<<<dim3(NTOK / 16, NBT), blk, 0, stream>>>(value, Wv, Vw, KEY_DIM);
  attn_kernel<<<dim3(NTOK / 16, NUM_HEADS, NBT), blk, 0, stream>>>(Qw, Kw, Vw, Aw);
  outproj_kernel<<<dim3(NTOK / 16, QUERY_DIM / 16, NBT), blk, 0, stream>>>(Aw, Wo, bo, out);
}